// CoCoNet_72249939853425
// MI455X (gfx1250) — compile-verified
//
#include <hip/hip_runtime.h>

typedef __attribute__((ext_vector_type(2))) float v2f;
typedef __attribute__((ext_vector_type(8))) float v8f;

#define NROI 148
#define WAVES 8
#define TILES_PER_WAVE 4  // 16-sample n-tiles per wave

// D = A(16 roi x K) * B(K x 16 n) + C, K padded to 8 = two K=4 WMMA slices.
// Shared B operand: F[n] = [X, Z0..Z4, 0, 0]
//   Wf_pad[r] = [Wf0..Wf5, 0, 0]        -> f_out
//   Wr_pad[r] = [0, Wr0..Wr4, 0, 0]     -> r_out
// FULL=true: all 16 rois of the tile are valid -> branch-free NT stores.
// FULL=false: tail tile (rois 144..147) -> guarded stores (1/10 of blocks).
template <bool FULL>
__global__ __launch_bounds__(WAVES * 32) void roi_linear_wmma(
    const float* __restrict__ X, const float* __restrict__ Z,
    const float* __restrict__ Wr, const float* __restrict__ br,
    const float* __restrict__ Wf, const float* __restrict__ bf,
    float* __restrict__ rOut, float* __restrict__ fOut, int Ntot,
    int roiBase0) {
  const int tid  = threadIdx.x;
  const int wave = tid >> 5;
  const int lane = tid & 31;
  const int ln   = lane & 15;  // column (B/C/D) or M row (A)
  const int hi   = lane >> 4;  // 0: K=0,1 half; 1: K=2,3 half

  const int rb  = roiBase0 + (int)blockIdx.y * 16;
  const int roi = rb + ln;
  const bool vr = FULL || (roi < NROI);

  // ---- A operands (weights), loop-invariant: loaded once per wave ----
  // A 16x4 layout: lane ln = M; VGPR0 holds K=2*hi, VGPR1 holds K=2*hi+1.
  v2f af0, af1, ar0, ar1;
  {
    const float* wf = Wf + (unsigned)(vr ? roi : 0) * 6u;
    const float* wr = Wr + (unsigned)(vr ? roi : 0) * 5u;
    const float g = vr ? 1.0f : 0.0f;
    const float wf0 = g * wf[0], wf1 = g * wf[1], wf2 = g * wf[2];
    const float wf3 = g * wf[3], wf4 = g * wf[4], wf5 = g * wf[5];
    const float wr0 = g * wr[0], wr1 = g * wr[1], wr2 = g * wr[2];
    const float wr3 = g * wr[3], wr4 = g * wr[4];
    af0.x = hi ? wf2 : wf0;  af0.y = hi ? wf3 : wf1;   // K = 2hi, 2hi+1
    af1.x = hi ? 0.f : wf4;  af1.y = hi ? 0.f : wf5;   // K = 4+2hi, 5+2hi
    ar0.x = hi ? wr1 : 0.f;  ar0.y = hi ? wr2 : wr0;   // Wr_pad shifted by 1
    ar1.x = hi ? 0.f : wr3;  ar1.y = hi ? 0.f : wr4;
  }

  // ---- biases as initial accumulators; per-lane row offsets (32-bit) ----
  // C/D layout: VGPR v -> row (v + 8*hi), column ln.
  v8f cf, cr;
  unsigned rowOff[8];
#pragma unroll
  for (int v = 0; v < 8; ++v) {
    const int row = rb + v + 8 * hi;
    const bool ok = FULL || (row < NROI);
    cf[v] = ok ? bf[row] : 0.f;
    cr[v] = ok ? br[row] : 0.f;
    rowOff[v] = (unsigned)row * (unsigned)Ntot + (unsigned)ln;
  }

  const unsigned nb0 =
      ((unsigned)blockIdx.x * WAVES + (unsigned)wave) * (16u * TILES_PER_WAVE);

#pragma unroll
  for (int t = 0; t < TILES_PER_WAVE; ++t) {
    const unsigned nb = nb0 + (unsigned)t * 16u;
    const unsigned n  = nb + (unsigned)ln;
    // Features for sample n (read by its lo & hi lane; tiny, L2-resident).
    const float x = X[n];
    const float* zp = Z + n * 5u;
    const float z0 = zp[0], z1 = zp[1], z2 = zp[2], z3 = zp[3], z4 = zp[4];
    // B 4x16: lane ln = column; VGPR0 = K=2hi, VGPR1 = K=2hi+1.
    v2f b0, b1;
    b0.x = hi ? z1 : x;    b0.y = hi ? z2 : z0;   // F[2hi],   F[2hi+1]
    b1.x = hi ? 0.f : z3;  b1.y = hi ? 0.f : z4;  // F[4+2hi], F[5+2hi]

    // Two independent accumulation chains, interleaved to hide the
    // WMMA->WMMA RAW hazard.
    v8f df = __builtin_amdgcn_wmma_f32_16x16x4_f32(false, af0, false, b0,
                                                   (short)0, cf, false, false);
    v8f dr = __builtin_amdgcn_wmma_f32_16x16x4_f32(false, ar0, false, b0,
                                                   (short)0, cr, false, false);
    df = __builtin_amdgcn_wmma_f32_16x16x4_f32(false, af1, false, b1,
                                               (short)0, df, false, false);
    dr = __builtin_amdgcn_wmma_f32_16x16x4_f32(false, ar1, false, b1,
                                               (short)0, dr, false, false);

    // Write-once output: non-temporal stores, 64B contiguous per half-wave
    // per VGPR. FULL path is branch-free (EXEC stays all-1s throughout).
#pragma unroll
    for (int v = 0; v < 8; ++v) {
      if (FULL || (rb + v + 8 * hi < NROI)) {
        const unsigned off = rowOff[v] + nb;
        __builtin_nontemporal_store(dr[v], rOut + off);
        __builtin_nontemporal_store(df[v], fOut + off);
      }
    }
  }
}

extern "C" void kernel_launch(void* const* d_in, const int* in_sizes, int n_in,
                              void* d_out, int out_size, void* d_ws,
                              size_t ws_size, hipStream_t stream) {
  (void)n_in; (void)out_size; (void)d_ws; (void)ws_size;
  const float* X  = (const float*)d_in[0];
  const float* Z  = (const float*)d_in[1];
  const float* Wr = (const float*)d_in[2];
  const float* br = (const float*)d_in[3];
  const float* Wf = (const float*)d_in[4];
  const float* bf = (const float*)d_in[5];

  const int Ntot = in_sizes[0];  // X is [N, 1]
  float* rOut = (float*)d_out;
  float* fOut = rOut + (size_t)NROI * (size_t)Ntot;

  const int samplesPerBlock = WAVES * 16 * TILES_PER_WAVE;  // 512
  const unsigned gx = (unsigned)(Ntot / samplesPerBlock);
  const int fullTiles = NROI / 16;       // 9 full 16-roi tiles
  const int tailBase  = fullTiles * 16;  // 144

  // Main body: branch-free store path for rois [0, 144).
  roi_linear_wmma<true><<<dim3(gx, (unsigned)fullTiles), WAVES * 32, 0,
                          stream>>>(X, Z, Wr, br, Wf, bf, rOut, fOut, Ntot, 0);
  // Tail: rois [144, 148) with guarded stores.
  if (NROI % 16) {
    roi_linear_wmma<false><<<dim3(gx, 1), WAVES * 32, 0, stream>>>(
        X, Z, Wr, br, Wf, bf, rOut, fOut, Ntot, tailBase);
  }
}